// AdaptiveNoiseMask_20770461843629
// MI455X (gfx1250) — compile-verified
//
#include <hip/hip_runtime.h>
#include <hip/hip_bf16.h>

// Problem constants (from reference setup_inputs)
#define C_DIM 1000      // classes
#define D_DIM 4096      // feature dim
#define BLOCK 256       // 8 wave32s per block
#define NVEC  ((D_DIM / 4) / BLOCK)   // 4 float4s per thread per array
#define K_NOISE_SCALE 0.1f
#define K_NOISE_RATIO 0.3f
#define K_MAX_NOISE   1.0f

typedef float v4f __attribute__((ext_vector_type(4)));
typedef __attribute__((address_space(3))) char lds_char;

// Generic (flat) LDS pointer -> 32-bit LDS byte address for async-to-LDS ops.
__device__ __forceinline__ unsigned lds_off(void* p) {
    return (unsigned)(unsigned long long)(lds_char*)p;
}

__global__ __launch_bounds__(BLOCK) void adaptive_noise_fused(
    const float* __restrict__ x,
    const float* __restrict__ mo,
    const float* __restrict__ ru,
    const float* __restrict__ ns,
    float* __restrict__ out)
{
    __shared__ __align__(16) float s_mo[1024];   // staged model_output row (1000 used)
    __shared__ float s_max[8];
    __shared__ float s_sum[8];

    const int b    = blockIdx.x;                 // one block per sample/row
    const int tid  = threadIdx.x;
    const int wid  = tid >> 5;
    const int lane = tid & 31;

    // ---- Fire async DMA: stage model_output[b, :] into LDS ----
    // Row stride = 1000 * 4 B = 4000 B (16B-divisible) => every row start is
    // 16B aligned; 250 lanes move one B128 each (exactly 4000 B, no OOB).
    {
        const unsigned long long row_base =
            (unsigned long long)(const void*)mo + (unsigned long long)b * (C_DIM * 4ull);
        if (tid < (C_DIM / 4)) {
            unsigned            laddr = lds_off(s_mo) + (unsigned)tid * 16u;
            unsigned long long  gaddr = row_base + (unsigned long long)tid * 16ull;
            asm volatile("global_load_async_to_lds_b128 %0, %1, off"
                         :: "v"(laddr), "v"(gaddr)
                         : "memory");
        }
    }

    // ---- Prefetch the streaming row data NOW (independent of softmax). ----
    // s_wait_asynccnt drains only ASYNCcnt, so these 12 B128 NT loads stay in
    // flight across the whole reduction and are first consumed at the end
    // (compiler inserts s_wait_loadcnt at use). Hides HBM latency of the
    // block's serial softmax phase.
    const size_t row = (size_t)b * D_DIM;
    const v4f* __restrict__ x4 = (const v4f*)(x  + row);
    const v4f* __restrict__ r4 = (const v4f*)(ru + row);
    const v4f* __restrict__ n4 = (const v4f*)(ns + row);
    v4f* __restrict__ o4       = (v4f*)(out + row);

    v4f xv[NVEC], rv[NVEC], nv[NVEC];
    #pragma unroll
    for (int k = 0; k < NVEC; ++k) {
        const int i = tid + k * BLOCK;
        xv[k] = __builtin_nontemporal_load(&x4[i]);
        rv[k] = __builtin_nontemporal_load(&r4[i]);
        nv[k] = __builtin_nontemporal_load(&n4[i]);
    }

#if __has_builtin(__builtin_amdgcn_s_wait_asynccnt)
    __builtin_amdgcn_s_wait_asynccnt(0);
#else
    asm volatile("s_wait_asynccnt 0" ::: "memory");
#endif
    __syncthreads();

    // ---- Row max (softmax stabilization) ----
    float lmax = -3.402823466e38f;
    for (int i = tid; i < C_DIM; i += BLOCK)
        lmax = fmaxf(lmax, s_mo[i]);
    #pragma unroll
    for (int off = 16; off > 0; off >>= 1)
        lmax = fmaxf(lmax, __shfl_xor(lmax, off, 32));
    if (lane == 0) s_max[wid] = lmax;
    __syncthreads();
    float rmax = s_max[0];
    #pragma unroll
    for (int w = 1; w < 8; ++w) rmax = fmaxf(rmax, s_max[w]);

    // ---- Sum of exp(logit - max); max softmax prob = exp(0)/sum = 1/sum ----
    float lsum = 0.0f;
    for (int i = tid; i < C_DIM; i += BLOCK)
        lsum += __expf(s_mo[i] - rmax);
    #pragma unroll
    for (int off = 16; off > 0; off >>= 1)
        lsum += __shfl_xor(lsum, off, 32);
    if (lane == 0) s_sum[wid] = lsum;
    __syncthreads();
    float rsum = 0.0f;
    #pragma unroll
    for (int w = 0; w < 8; ++w) rsum += s_sum[w];

    const float confidence = 1.0f / rsum;
    const float scale = fminf(K_NOISE_SCALE * (1.0f + confidence), K_MAX_NOISE);

    // ---- Consume prefetched data: out = x + (rand_u < ratio ? noise*scale : 0)
    #pragma unroll
    for (int k = 0; k < NVEC; ++k) {
        const int i = tid + k * BLOCK;
        v4f ov;
        ov.x = xv[k].x + ((rv[k].x < K_NOISE_RATIO) ? nv[k].x * scale : 0.0f);
        ov.y = xv[k].y + ((rv[k].y < K_NOISE_RATIO) ? nv[k].y * scale : 0.0f);
        ov.z = xv[k].z + ((rv[k].z < K_NOISE_RATIO) ? nv[k].z * scale : 0.0f);
        ov.w = xv[k].w + ((rv[k].w < K_NOISE_RATIO) ? nv[k].w * scale : 0.0f);
        __builtin_nontemporal_store(ov, &o4[i]);
    }
}

extern "C" void kernel_launch(void* const* d_in, const int* in_sizes, int n_in,
                              void* d_out, int out_size, void* d_ws, size_t ws_size,
                              hipStream_t stream) {
    const float* x  = (const float*)d_in[0];   // [B, D] f32
    const float* mo = (const float*)d_in[1];   // [B, C] f32
    const float* ru = (const float*)d_in[2];   // [B, D] f32
    const float* ns = (const float*)d_in[3];   // [B, D] f32
    float* out = (float*)d_out;                // [B, D] f32

    const int B = in_sizes[1] / C_DIM;         // 4096
    adaptive_noise_fused<<<dim3(B), dim3(BLOCK), 0, stream>>>(x, mo, ru, ns, out);
}